// SANLayer_27779848470631
// MI455X (gfx1250) — compile-verified
//
#include <hip/hip_runtime.h>
#include <hip/hip_bf16.h>

// ---------------- problem constants ----------------
constexpr int N  = 50000;
constexpr int E  = 800000;
constexpr int D  = 64;       // IN_DIM == OUT_DIM
constexpr int H  = 8;
constexpr int HD = 8;
constexpr float INV_SQRT_HD = 0.35355339059327373f;

// ---------------- WMMA helpers (fp32, 16x16x4) ----------------
typedef float v2f __attribute__((ext_vector_type(2)));
typedef float v8f __attribute__((ext_vector_type(8)));

// ISA layout (CDNA5 ISA):
//   A 16x4 f32: lanes 0-15 rows M, VGPR0={K0|K2}, VGPR1={K1|K3}  -> float2 per lane
//   B 4x16:     lanes 0-15 cols N, same K striping across lane halves
//   C/D v8f:    VGPR r -> M=r (lanes 0-15) / M=8+r (lanes 16-31), N=lane&15
__device__ __forceinline__ v8f wmma_k_accum(const float* __restrict__ A, int lda,
                                            const float* __restrict__ B, int ldb,
                                            int col0, int K, v8f acc) {
    const int lane = threadIdx.x & 31;
    const int hf   = lane >> 4;     // lane half
    const int m    = lane & 15;     // row (A) / col (B) within tile
    const float* arow = A + m * lda + 2 * hf;
    for (int k0 = 0; k0 < K; k0 += 4) {
        v2f a, b;
        a.x = arow[k0];
        a.y = arow[k0 + 1];
        const int kb = k0 + 2 * hf;
        b.x = B[kb * ldb + col0 + m];
        b.y = B[(kb + 1) * ldb + col0 + m];
        acc = __builtin_amdgcn_wmma_f32_16x16x4_f32(
            false, a, false, b, (short)0, acc, false, false);
    }
    return acc;
}

__device__ __forceinline__ void wmma_store_tile(float* __restrict__ Y, int ldy,
                                                int row0, int col0, v8f c) {
    const int lane = threadIdx.x & 31;
    const int hf   = lane >> 4;
    const int m    = lane & 15;
    for (int r = 0; r < 8; ++r)
        Y[(size_t)(row0 + r + 8 * hf) * ldy + col0 + m] = c[r];
}

// ---------------- kernels ----------------
__global__ void zero_kernel(float* p, int n) {
    int i = blockIdx.x * blockDim.x + threadIdx.x;
    int stride = gridDim.x * blockDim.x;
    for (; i < n; i += stride) p[i] = 0.0f;
}

// Persistent-wave GEMM: Y[rows x 64] = A[rows x 64] @ B[64 x 64].
// Each wave owns one 16-col tile (ct = wave & 3): its 16 B-fragments are
// loaded ONCE into registers, then row tiles are streamed with exactly one
// global_load_b64 + one v_wmma per K-step (feeds the matrix pipe at the
// HBM streaming rate; B never re-read).
__global__ void gemm64_persist_kernel(const float* __restrict__ A,
                                      const float* __restrict__ B,
                                      float* __restrict__ Y, int rows) {
    const int gw     = (blockIdx.x * blockDim.x + threadIdx.x) >> 5;
    const int nwaves = (gridDim.x * blockDim.x) >> 5;
    const int ct     = gw & 3;          // column tile 0..3
    const int wslot  = gw >> 2;         // row-stripe slot
    const int nslots = nwaves >> 2;
    const int ntiles = rows >> 4;
    const int lane = threadIdx.x & 31;
    const int hf   = lane >> 4;
    const int m    = lane & 15;

    // load the 16 B fragments for this column tile once (32 VGPRs)
    v2f bf[16];
    #pragma unroll
    for (int s = 0; s < 16; ++s) {
        const int kb = 4 * s + 2 * hf;
        bf[s].x = B[kb * D + ct * 16 + m];
        bf[s].y = B[(kb + 1) * D + ct * 16 + m];
    }

    for (int rt = wslot; rt < ntiles; rt += nslots) {
        const float* arow = A + (size_t)rt * 16 * D + m * D + 2 * hf;
        // speculative prefetch of the next row tile this wave will touch
        __builtin_prefetch(arow + (size_t)nslots * 16 * D, 0, 1);
        v8f acc = {};
        #pragma unroll
        for (int s = 0; s < 16; ++s) {
            v2f a;
            a.x = arow[4 * s];
            a.y = arow[4 * s + 1];
            acc = __builtin_amdgcn_wmma_f32_16x16x4_f32(
                false, a, false, bf[s], (short)0, acc, false, false);
        }
        wmma_store_tile(Y, D, rt * 16, ct * 16, acc);
    }
}

// one thread per (edge, head)
__global__ void edge_attn_kernel(const int* __restrict__ ei,
                                 const float* __restrict__ Qh, const float* __restrict__ Kh,
                                 const float* __restrict__ Vh, const float* __restrict__ Eh,
                                 float* __restrict__ wV, float* __restrict__ Z) {
    const int t = blockIdx.x * blockDim.x + threadIdx.x;
    if (t >= E * H) return;
    const int e = t >> 3;
    const int h = t & 7;
    const int src = ei[e];
    const int dst = ei[E + e];
    const float* kp = Kh + (size_t)src * D + h * HD;
    const float* qp = Qh + (size_t)dst * D + h * HD;
    const float* ep = Eh + (size_t)e * D + h * HD;
    float s = 0.0f;
    #pragma unroll
    for (int d = 0; d < HD; ++d) s += kp[d] * qp[d] * ep[d];
    s *= INV_SQRT_HD;
    s = fminf(5.0f, fmaxf(-5.0f, s));
    s = __expf(s);
    const float* vp = Vh + (size_t)src * D + h * HD;
    float* wp = wV + (size_t)dst * D + h * HD;
    #pragma unroll
    for (int d = 0; d < HD; ++d) atomicAdd(&wp[d], vp[d] * s);
    atomicAdd(&Z[dst * H + h], s);
}

// wV <- wV / (Z + 1e-6), in place
__global__ void norm_attn_kernel(float* __restrict__ wV, const float* __restrict__ Z) {
    const int i = blockIdx.x * blockDim.x + threadIdx.x;
    if (i >= N * D) return;
    const int n = i >> 6;
    const int h = (i >> 3) & 7;
    wV[i] = wV[i] / (Z[n * H + h] + 1e-6f);
}

// h1 = x + (h_attn @ O_w + O_b); accumulate per-column sum / sumsq for BN1
__global__ void oproj_kernel(const float* __restrict__ hA, const float* __restrict__ Ow,
                             const float* __restrict__ Ob, const float* __restrict__ x,
                             float* __restrict__ h1, float* __restrict__ sum1,
                             float* __restrict__ sq1) {
    const int wave   = (blockIdx.x * blockDim.x + threadIdx.x) >> 5;
    const int ntiles = N >> 4;
    if (wave >= ntiles) return;               // wave-uniform: EXEC all-1 for WMMA
    const int row0 = wave << 4;
    const float* Atile = hA + (size_t)row0 * D;
    const int lane = threadIdx.x & 31;
    const int hf   = lane >> 4;
    const int m    = lane & 15;
    for (int ct = 0; ct < 4; ++ct) {
        v8f acc = {};
        acc = wmma_k_accum(Atile, D, Ow, D, ct * 16, D, acc);
        const int col  = ct * 16 + m;
        const float bias = Ob[col];
        for (int r = 0; r < 8; ++r) {
            const int row = row0 + r + 8 * hf;
            const float val = acc[r] + bias + x[(size_t)row * D + col];
            h1[(size_t)row * D + col] = val;
            atomicAdd(&sum1[col], val);
            atomicAdd(&sq1[col], val * val);
        }
    }
}

// finalize BN stats -> per-column scale/shift
__global__ void bnstats_kernel(const float* __restrict__ sum, const float* __restrict__ sq,
                               const float* __restrict__ g, const float* __restrict__ b,
                               float* __restrict__ scale, float* __restrict__ shift) {
    const int c = threadIdx.x;
    if (c < D) {
        const float mu  = sum[c] * (1.0f / (float)N);
        const float var = sq[c] * (1.0f / (float)N) - mu * mu;
        const float s   = g[c] * rsqrtf(var + 1e-5f);
        scale[c] = s;
        shift[c] = b[c] - mu * s;
    }
}

// FFN per 16-row tile (1 wave per block): hbn = bn1(h1) staged in LDS,
// act = relu(hbn@f1 + b1) staged in LDS, hp2 = hbn + act@f2 + b2; BN2 stats.
__global__ void ffn_kernel(const float* __restrict__ h1,
                           const float* __restrict__ scale1, const float* __restrict__ shift1,
                           const float* __restrict__ f1w, const float* __restrict__ f1b,
                           const float* __restrict__ f2w, const float* __restrict__ f2b,
                           float* __restrict__ hp2, float* __restrict__ sum2,
                           float* __restrict__ sq2) {
    __shared__ float shh[16 * 64];    // BN1-normalized input tile
    __shared__ float sha[16 * 128];   // relu activations
    const int row0 = blockIdx.x << 4;
    const int lane = threadIdx.x;     // blockDim == 32
    #pragma unroll
    for (int i = 0; i < 32; ++i) {
        const int idx = i * 32 + lane;
        const int row = idx >> 6;
        const int col = idx & 63;
        const float v = h1[(size_t)(row0 + row) * D + col];
        shh[idx] = v * scale1[col] + shift1[col];
    }
    __syncthreads();
    const int hf = lane >> 4;
    const int m  = lane & 15;
    // GEMM1 (K=64 -> 128 cols) + bias + relu
    for (int ct = 0; ct < 8; ++ct) {
        const int col = ct * 16 + m;
        v8f acc;
        const float b1 = f1b[col];
        for (int r = 0; r < 8; ++r) acc[r] = b1;
        acc = wmma_k_accum(shh, 64, f1w, 128, ct * 16, 64, acc);
        for (int r = 0; r < 8; ++r) {
            const float v = acc[r] > 0.0f ? acc[r] : 0.0f;
            sha[(r + 8 * hf) * 128 + col] = v;
        }
    }
    __syncthreads();
    // GEMM2 (K=128 -> 64 cols) + bias + residual; BN2 stats
    for (int ct = 0; ct < 4; ++ct) {
        const int col = ct * 16 + m;
        v8f acc;
        const float b2 = f2b[col];
        for (int r = 0; r < 8; ++r) acc[r] = b2;
        acc = wmma_k_accum(sha, 128, f2w, 64, ct * 16, 128, acc);
        for (int r = 0; r < 8; ++r) {
            const int row = r + 8 * hf;
            const float val = acc[r] + shh[row * 64 + col];
            hp2[(size_t)(row0 + row) * D + col] = val;
            atomicAdd(&sum2[col], val);
            atomicAdd(&sq2[col], val * val);
        }
    }
}

// out = bn2(hp2)
__global__ void bn_apply_kernel(const float* __restrict__ hp2,
                                const float* __restrict__ scale2,
                                const float* __restrict__ shift2,
                                float* __restrict__ out) {
    const int i = blockIdx.x * blockDim.x + threadIdx.x;
    if (i >= N * D) return;
    const int col = i & 63;
    out[i] = hp2[i] * scale2[col] + shift2[col];
}

// ---------------- host launch ----------------
extern "C" void kernel_launch(void* const* d_in, const int* in_sizes, int n_in,
                              void* d_out, int out_size, void* d_ws, size_t ws_size,
                              hipStream_t stream) {
    const float* x         = (const float*)d_in[0];
    const float* edge_attr = (const float*)d_in[1];
    const int*   edge_idx  = (const int*)  d_in[2];
    const float* Wq        = (const float*)d_in[3];
    const float* Wk        = (const float*)d_in[4];
    const float* We        = (const float*)d_in[5];
    const float* Wv        = (const float*)d_in[6];
    const float* Ow        = (const float*)d_in[7];
    const float* Ob        = (const float*)d_in[8];
    const float* f1w       = (const float*)d_in[9];
    const float* f1b       = (const float*)d_in[10];
    const float* f2w       = (const float*)d_in[11];
    const float* f2b       = (const float*)d_in[12];
    const float* bn1g      = (const float*)d_in[13];
    const float* bn1b      = (const float*)d_in[14];
    const float* bn2g      = (const float*)d_in[15];
    const float* bn2b      = (const float*)d_in[16];
    float* out = (float*)d_out;

    // workspace layout (floats)
    float* ws = (float*)d_ws;
    const size_t ND = (size_t)N * D;
    float* Qh    = ws;                      // N*64
    float* Kh    = Qh + ND;                 // N*64
    float* Vh    = Kh + ND;                 // N*64
    float* wV    = Vh + ND;                 // N*64 (accumulator, then h_attn)
    float* Z     = wV + ND;                 // N*8
    float* h1    = Z  + (size_t)N * H;      // N*64
    float* hp2   = h1 + ND;                 // N*64
    float* stats = hp2 + ND;                // 512 floats
    float* Eh    = stats + 512;             // E*64
    float* sum1   = stats + 0;
    float* sq1    = stats + 64;
    float* sum2   = stats + 128;
    float* sq2    = stats + 192;
    float* scale1 = stats + 256;
    float* shift1 = stats + 320;
    float* scale2 = stats + 384;
    float* shift2 = stats + 448;

    // 1) zero accumulators (wV + Z are contiguous) and stats
    {
        const int nz = (int)(ND + (size_t)N * H);
        zero_kernel<<<(nz + 255) / 256, 256, 0, stream>>>(wV, nz);
        zero_kernel<<<1, 256, 0, stream>>>(stats, 512);
    }

    // 2) projections via persistent-wave WMMA GEMMs (B fragments in registers)
    {
        gemm64_persist_kernel<<<512,  256, 0, stream>>>(x,         Wq, Qh, N);
        gemm64_persist_kernel<<<512,  256, 0, stream>>>(x,         Wk, Kh, N);
        gemm64_persist_kernel<<<512,  256, 0, stream>>>(x,         Wv, Vh, N);
        gemm64_persist_kernel<<<2048, 256, 0, stream>>>(edge_attr, We, Eh, E);
    }

    // 3) edge attention scatter
    {
        const int nt = E * H;
        edge_attn_kernel<<<(nt + 255) / 256, 256, 0, stream>>>(edge_idx, Qh, Kh, Vh, Eh, wV, Z);
    }

    // 4) normalize: h_attn = wV / (Z + 1e-6)
    norm_attn_kernel<<<((int)ND + 255) / 256, 256, 0, stream>>>(wV, Z);

    // 5) O projection + residual + BN1 stats
    {
        const int waves = N / 16;
        oproj_kernel<<<(waves + 7) / 8, 256, 0, stream>>>(wV, Ow, Ob, x, h1, sum1, sq1);
    }

    // 6) BN1 finalize
    bnstats_kernel<<<1, 64, 0, stream>>>(sum1, sq1, bn1g, bn1b, scale1, shift1);

    // 7) FFN + residual + BN2 stats (1 wave / 16-row tile)
    ffn_kernel<<<N / 16, 32, 0, stream>>>(h1, scale1, shift1, f1w, f1b, f2w, f2b,
                                          hp2, sum2, sq2);

    // 8) BN2 finalize
    bnstats_kernel<<<1, 64, 0, stream>>>(sum2, sq2, bn2g, bn2b, scale2, shift2);

    // 9) apply BN2 -> output
    bn_apply_kernel<<<((int)ND + 255) / 256, 256, 0, stream>>>(hp2, scale2, shift2, out);

    (void)in_sizes; (void)n_in; (void)out_size; (void)ws_size;
}